// NeuralODEAdjoint_69681549410552
// MI455X (gfx1250) — compile-verified
//
#include <hip/hip_runtime.h>
#include <hip/hip_bf16.h>

typedef __attribute__((ext_vector_type(2))) float v2f;
typedef __attribute__((ext_vector_type(8))) float v8f;
typedef __attribute__((ext_vector_type(4))) unsigned int v4u;
typedef __attribute__((ext_vector_type(4))) int v4i;
typedef __attribute__((ext_vector_type(8))) int v8i;

namespace {
constexpr int kN = 64;     // state dim
constexpr int kH = 512;    // hidden dim
constexpr int kP = 32;     // theta dim
constexpr int kM = 32;     // obs dim
constexpr int kT = 512;    // timesteps
constexpr int kNF = 8;     // n_forecasts
constexpr int kPts = kT * kNF;                    // 4096 integration points
constexpr float kDt = 51.2f / (float)(kPts - 1);  // T*OBS_DT/(n_points-1)
}  // namespace

#if defined(__has_builtin)
#if __has_builtin(__builtin_amdgcn_tensor_load_to_lds)
#define HAVE_TDM 1
#else
#define HAVE_TDM 0
#endif
#else
#define HAVE_TDM 0
#endif

__device__ __forceinline__ v8f wmma4(v2f a, v2f b, v8f c) {
  // D = A(16x4,f32) * B(4x16,f32) + C(16x16,f32)
  return __builtin_amdgcn_wmma_f32_16x16x4_f32(false, a, false, b, (short)0, c,
                                               false, false);
}

// Branchless tanh on the critical path: hardware V_TANH_F32 if exposed,
// else 2 transcendentals (v_exp_f32 + v_rcp_f32). Saturates to +/-1.
__device__ __forceinline__ float fast_tanh(float x) {
#if defined(__has_builtin) && __has_builtin(__builtin_amdgcn_tanhf)
  return __builtin_amdgcn_tanhf(x);
#else
  const float e2x = __builtin_amdgcn_exp2f(x * 2.88539008177793f);  // 2*log2(e)
  return 1.0f - 2.0f * __builtin_amdgcn_rcpf(e2x + 1.0f);
#endif
}

#if HAVE_TDM
// Tensor Data Mover: DMA a row-major 2D f32 tile (rows x cols, row stride in
// elements) from global memory into LDS. D# built per CDNA5 ISA ch.8.
// Wave-level op (EXEC ignored) tracked with TENSORcnt.
__device__ __forceinline__ void tdm_load_2d(const void* gptr, void* lptr,
                                            unsigned rows, unsigned cols,
                                            unsigned row_stride_elems) {
  const unsigned long long ga = (unsigned long long)(uintptr_t)gptr;
  // flat shared address: [63:32]=shared aperture, [31:0]=LDS byte offset
  const unsigned lds_off = (unsigned)(uintptr_t)lptr;
  v4u g0;
  g0.x = 0x1u;                                   // count=1, user descriptor
  g0.y = lds_off;                                // lds_addr [63:32]
  g0.z = (unsigned)(ga & 0xFFFFFFFFu);           // global_addr [95:64]
  g0.w = (unsigned)((ga >> 32) & 0x01FFFFFFu)    // global_addr [120:96]
         | (2u << 30);                           // type=2 ("image")
  v8i g1;
  g1[0] = (int)(2u << 16);                                   // data_size = 4B
  g1[1] = (int)((cols & 0xFFFFu) << 16);                     // tensor_dim0 lo
  g1[2] = (int)(((cols >> 16) & 0xFFFFu) | ((rows & 0xFFFFu) << 16));
  g1[3] = (int)(((rows >> 16) & 0xFFFFu) | ((cols & 0xFFFFu) << 16));  // tile_dim0
  g1[4] = (int)(rows & 0xFFFFu);                             // tile_dim1
  g1[5] = (int)row_stride_elems;                             // dim0_stride lo32
  g1[6] = 0;
  g1[7] = 0;
  const v4i gz = {0, 0, 0, 0};
#if __clang_major__ >= 23
  const v8i gz8 = {0, 0, 0, 0, 0, 0, 0, 0};
  __builtin_amdgcn_tensor_load_to_lds(g0, g1, gz, gz, gz8, 0);
#else
  __builtin_amdgcn_tensor_load_to_lds(g0, g1, gz, gz, 0);
#endif
}
#endif  // HAVE_TDM

// ---------------------------------------------------------------------------
// Kernel A: persistent single-workgroup RK4 scan.
// 512 threads = 16 waves on one WGP. Weights live in LDS (320KB/WGP on CDNA5),
// staged once by the Tensor Data Mover. Each f(x):
//   layer1: h = tanh(x@W1 + theta_h + b1)   (1x64 @ 64x512) via f32 WMMA
//   layer2: f = h@W2 + b2                   (1x512 @ 512x64) via f32 WMMA
// The 1xK vector sits in row 0 of the 16-row A fragment; lanes carrying
// rows 1..15 read from a zeroed LDS region (no EXEC masking, no reg fills).
// Layer-2 partial reduction is fused into the RK4 stage update (3 barriers
// per stage instead of 4).
// ---------------------------------------------------------------------------
__global__ __launch_bounds__(512) void ode_rk4_kernel(
    const float* __restrict__ X0, const float* __restrict__ theta0,
    const float* __restrict__ W1, const float* __restrict__ Wth,
    const float* __restrict__ b1, const float* __restrict__ W2,
    const float* __restrict__ b2, float* __restrict__ Xpred) {
  extern __shared__ float lds[];
  float* sW1 = lds;            // 64*512
  float* sW2 = sW1 + kN * kH;  // 512*64
  float* sTh = sW2 + kH * kN;  // 512 : theta@Wth + b1 (precomputed)
  float* sHb = sTh + kH;       // 512 : tanh hidden activations
  float* sB2 = sHb + kH;       // 64
  float* sX = sB2 + kN;        // 64 : current state
  float* sXt = sX + kN;        // 64 : stage input
  float* sK = sXt + kN;        // 64 : k1 + 2k2 + 2k3 accumulator
  float* sP = sK + kN;         // 4*64 : layer2 K-group partials
  float* sZ = sP + 4 * kN;     // 512 : zero pad source for A rows 1..15

  const int tid = threadIdx.x;
  const int wave = tid >> 5;
  const int lane = tid & 31;
  const int half = lane >> 4;  // 0: K pair {0,1}, 1: K pair {2,3}
  const int lan16 = lane & 15;

  // ---- stage weights into LDS (persistent for 16380 matvecs) -------------
#if HAVE_TDM
  if (wave == 0) {  // wave-uniform: TDM issues per-wave regardless of EXEC
    tdm_load_2d(W1, sW1, kN, kH, kH);
    tdm_load_2d(W2, sW2, kH, kN, kN);
  }
#else
  for (int i = tid; i < kN * kH; i += 512) sW1[i] = W1[i];
  for (int i = tid; i < kH * kN; i += 512) sW2[i] = W2[i];
#endif
  // theta_h = theta0 @ Wth + b1  (tiny: 512 outputs x 32 MACs)
  {
    float acc = b1[tid];
    for (int i = 0; i < kP; ++i) acc += theta0[i] * Wth[i * kH + tid];
    sTh[tid] = acc;
    sZ[tid] = 0.0f;
  }
  if (tid < kN) {
    sB2[tid] = b2[tid];
    const float x0 = X0[tid];
    sX[tid] = x0;
    Xpred[tid] = x0;  // traj row 0 is X0
  }
#if HAVE_TDM
#if __has_builtin(__builtin_amdgcn_s_wait_tensorcnt)
  if (wave == 0) __builtin_amdgcn_s_wait_tensorcnt(0);
#endif
#endif
  __syncthreads();

  // A-fragment base for layer 2 is fixed (sHb); select once.
  const float* __restrict__ abH = (lan16 == 0) ? sHb : sZ;

  // f(xin): leaves layer-2 K-group partials in sP (reduced by caller).
  auto feval = [&](const float* __restrict__ xin) {
    // Lanes with lan16!=0 fetch zeros from sZ -> unconditional ds_load_b64.
    const float* __restrict__ abX = (lan16 == 0) ? xin : sZ;
    // ---- layer 1: each wave produces 32 hidden columns (2 x 16-col tiles)
#pragma unroll
    for (int t = 0; t < 2; ++t) {
      const int col = wave * 32 + t * 16 + lan16;
      v8f acc = {};
#pragma unroll
      for (int kk = 0; kk < kN / 4; ++kk) {
        const int kb = kk * 4 + 2 * half;  // even -> 8B aligned
        const v2f a = *(const v2f*)(abX + kb);
        v2f b;
        b.x = sW1[(kb + 0) * kH + col];
        b.y = sW1[(kb + 1) * kH + col];
        acc = wmma4(a, b, acc);
      }
      if (half == 0) sHb[col] = fast_tanh(acc[0] + sTh[col]);  // D row 0
    }
    __syncthreads();
    // ---- layer 2: wave -> (N-tile = wave&3, K-group = wave>>2)
    {
      const int nt = wave & 3;
      const int kg = wave >> 2;
      const int col = nt * 16 + lan16;
      v8f acc = {};
#pragma unroll
      for (int kk = 0; kk < 32; ++kk) {
        const int kb = kg * 128 + kk * 4 + 2 * half;
        const v2f a = *(const v2f*)(abH + kb);
        v2f b;
        b.x = sW2[(kb + 0) * kN + col];
        b.y = sW2[(kb + 1) * kN + col];
        acc = wmma4(a, b, acc);
      }
      if (half == 0) sP[kg * kN + col] = acc[0];
    }
    __syncthreads();
  };
  // Fused partial reduction: f[t] for t < 64.
  auto fsum = [&](int t) {
    return sP[t] + sP[kN + t] + sP[2 * kN + t] + sP[3 * kN + t] + sB2[t];
  };

  for (int s = 1; s < kPts; ++s) {
    feval(sX);  // k1 partials
    if (tid < kN) {
      const float k1 = fsum(tid);
      sK[tid] = k1;
      sXt[tid] = sX[tid] + 0.5f * kDt * k1;
    }
    __syncthreads();
    feval(sXt);  // k2
    if (tid < kN) {
      const float k2 = fsum(tid);
      sK[tid] += 2.0f * k2;
      sXt[tid] = sX[tid] + 0.5f * kDt * k2;
    }
    __syncthreads();
    feval(sXt);  // k3
    if (tid < kN) {
      const float k3 = fsum(tid);
      sK[tid] += 2.0f * k3;
      sXt[tid] = sX[tid] + kDt * k3;
    }
    __syncthreads();
    feval(sXt);  // k4
    if (tid < kN) {
      const float xn = sX[tid] + (kDt / 6.0f) * (sK[tid] + fsum(tid));
      sX[tid] = xn;
      if ((s & 7) == 0) Xpred[(s >> 3) * kN + tid] = xn;  // traj[::8]
    }
    __syncthreads();
  }
}

// ---------------------------------------------------------------------------
// Kernel B: Y_hat = X_pred @ H + bH  (512x64 @ 64x32) as 64 WMMA tiles,
// fused with deterministic per-block SSE partial (shfl-xor tree, lane 0 writes).
// One wave (32 threads) per 16x16 output tile.
// ---------------------------------------------------------------------------
__global__ __launch_bounds__(32) void yhat_loss_kernel(
    const float* __restrict__ Xpred, const float* __restrict__ Hm,
    const float* __restrict__ bH, const float* __restrict__ obs,
    float* __restrict__ Yhat, float* __restrict__ partial) {
  const int lane = threadIdx.x;
  const int half = lane >> 4;
  const int lan16 = lane & 15;
  const int mt = blockIdx.x >> 1;  // 0..31 row tiles
  const int nt = blockIdx.x & 1;   // 0..1 col tiles
  const int col = nt * 16 + lan16;
  const int arow = mt * 16 + lan16;

  v8f acc = {};
#pragma unroll
  for (int kk = 0; kk < kN / 4; ++kk) {
    const int kb = kk * 4 + 2 * half;
    v2f a, b;
    a.x = Xpred[arow * kN + kb + 0];
    a.y = Xpred[arow * kN + kb + 1];
    b.x = Hm[(kb + 0) * kM + col];
    b.y = Hm[(kb + 1) * kM + col];
    acc = wmma4(a, b, acc);
  }

  float local = 0.0f;
  const float bias = bH[col];
#pragma unroll
  for (int i = 0; i < 8; ++i) {
    const int row = mt * 16 + half * 8 + i;  // D layout: vgpr i -> M=i / M=8+i
    const float y = acc[i] + bias;
    Yhat[row * kM + col] = y;
    const float d = y - obs[row * kM + col];
    local += d * d;
  }
  // deterministic wave32 reduction
  for (int off = 16; off > 0; off >>= 1) local += __shfl_xor(local, off, 32);
  if (lane == 0) partial[blockIdx.x] = local;
}

// Kernel C: deterministic fixed-order final reduction -> scalar loss.
__global__ void loss_finalize_kernel(const float* __restrict__ partial,
                                     float* __restrict__ loss) {
  float s = 0.0f;
  for (int i = 0; i < 64; ++i) s += partial[i];
  *loss = s / (float)(kT * kM);
}

extern "C" void kernel_launch(void* const* d_in, const int* in_sizes, int n_in,
                              void* d_out, int out_size, void* d_ws,
                              size_t ws_size, hipStream_t stream) {
  (void)in_sizes; (void)n_in; (void)out_size; (void)ws_size;
  const float* X0  = (const float*)d_in[0];
  const float* th0 = (const float*)d_in[1];
  const float* obs = (const float*)d_in[2];
  const float* W1  = (const float*)d_in[3];
  const float* Wth = (const float*)d_in[4];
  const float* b1  = (const float*)d_in[5];
  const float* W2  = (const float*)d_in[6];
  const float* b2  = (const float*)d_in[7];
  const float* Hm  = (const float*)d_in[8];
  const float* bH  = (const float*)d_in[9];
  // d_in[10] = n_forecasts (compile-time constant 8 here)

  float* out = (float*)d_out;
  float* Xpred = out;                     // 512*64
  float* Yhat = out + kT * kN;            // 512*32
  float* loss = out + kT * kN + kT * kM;  // 1
  float* partial = (float*)d_ws;          // 64 floats of scratch

  // LDS: W1 + W2 + theta_h + hbuf + b2/x/xt/k + partials + zero pad
  constexpr size_t kSmemFloats = (size_t)kN * kH + (size_t)kH * kN + kH + kH +
                                 kN * 3 + 4 * kN + kH;
  constexpr size_t kSmemBytes = kSmemFloats * sizeof(float);  // ~264 KB < 320 KB

  ode_rk4_kernel<<<1, 512, kSmemBytes, stream>>>(X0, th0, W1, Wth, b1, W2, b2,
                                                 Xpred);
  yhat_loss_kernel<<<64, 32, 0, stream>>>(Xpred, Hm, bH, obs, Yhat, partial);
  loss_finalize_kernel<<<1, 1, 0, stream>>>(partial, loss);
}